// GAT4Rec_16234976379092
// MI455X (gfx1250) — compile-verified
//
#include <hip/hip_runtime.h>
#include <math.h>
#include <stdint.h>

#define DIM        64
#define HALF_DIM   32
#define NNB        50
#define ROWS       64     // 50 neighbors + 1 target + zero padding
#define APITCH     65     // 64 dwords data + 1 dword TDM pad -> (r+k)%64 banks
#define WPITCH     33
#define NWPITCH    33     // gcd(33,64)=1 -> conflict-free row- and col-indexed

typedef __attribute__((ext_vector_type(2))) float v2f;
typedef __attribute__((ext_vector_type(8))) float v8f;
typedef __attribute__((ext_vector_type(4))) unsigned int u32x4;
typedef __attribute__((ext_vector_type(8))) unsigned int u32x8;

__global__ __launch_bounds__(128)
void gat4rec_fused(const int* __restrict__ u,
                   const int* __restrict__ itm,
                   const int* __restrict__ neighbors,
                   const float* __restrict__ user_emb,
                   const float* __restrict__ entity_emb,
                   const float* __restrict__ Wg,
                   const float* __restrict__ ag,
                   float* __restrict__ out,
                   int n_entities)
{
    __shared__ float sA[ROWS * APITCH];    // renormed entity rows (A matrix)
    __shared__ float sW[DIM * WPITCH];     // W (64x32), K-major
    __shared__ float sNW[ROWS * NWPITCH];  // NW = A @ W  (64x32)
    __shared__ float sa[DIM];              // attention vector a
    __shared__ float sE[ROWS];             // logits -> aij
    __shared__ float sH[HALF_DIM];         // attention-pooled h
    __shared__ float sPart[ROWS * 2];      // per-half-row sumsq partials
    __shared__ float sScale[ROWS];         // renorm scales
    __shared__ float sHP[4 * HALF_DIM];    // h partials (one per wave)

    const int b    = blockIdx.x;
    const int tid  = threadIdx.x;
    const int lane = tid & 31;
    const int wave = tid >> 5;

    // ---- wave 0: TDM gather of 51 entity rows into sA (pitch-65 via pad) ----
    if (wave == 0) {
        const int* nb = neighbors + b * NNB;            // wave-uniform -> s_loads
        const int  tgt = itm[b];

        // prefetch the user row while we wait on the gather
        const long uid = u[b];
        __builtin_prefetch(user_emb + uid * DIM, 0, 0);
        __builtin_prefetch(user_emb + uid * DIM + 32, 0, 0);

        const unsigned long long gaddr = (unsigned long long)(uintptr_t)entity_emb;
        const unsigned int lds0 = (unsigned int)(uintptr_t)sA;  // LDS byte offset

        // D# group1 (shared by all descriptors except tile_dim1)
        // w0: data_size=2(4B) | pad_enable | pad_interval=5(64 dwords) | pad_amount=0(1 dword)
        u32x8 g1;
        g1[0] = (2u << 16) | (1u << 20) | (5u << 22);
        g1[1] = (unsigned)(DIM << 16);                  // tensor_dim0[15:0] @ bits 63:48
        g1[2] = ((unsigned)n_entities & 0xFFFFu) << 16; // dim0 hi(=0) | tensor_dim1 lo16
        g1[3] = (((unsigned)n_entities >> 16) & 0xFFFFu) | ((unsigned)DIM << 16); // dim1 hi | tile_dim0
        g1[4] = 0;                                      // tile_dim1 per descriptor
        g1[5] = (unsigned)DIM;                          // tensor_dim0_stride = 64 elements
        g1[6] = 0;
        g1[7] = 0;

        u32x4 g0;
        g0[0] = 0xC0000001u;                            // count=1 | idx32 | gather_mode
        g0[2] = (unsigned)(gaddr & 0xFFFFFFFFull);
        g0[3] = ((unsigned)(gaddr >> 32) & 0x01FFFFFFu) | 0x80000000u; // addr hi | type=2

        for (int d = 0; d < 7; ++d) {
            const int base = d * 8;
            const int nvalid = (d < 6) ? 8 : 3;         // 6*8 + 3 = 51 rows
            unsigned idxs[8];
            #pragma unroll
            for (int j = 0; j < 8; ++j) {
                const int r = base + j;
                idxs[j] = (r < NNB) ? (unsigned)nb[r]
                        : (r == NNB) ? (unsigned)tgt
                                     : (unsigned)tgt;   // padding (ignored, non-decreasing-ish)
            }
            u32x4 g2, g3;
            g2[0] = idxs[0]; g2[1] = idxs[1]; g2[2] = idxs[2]; g2[3] = idxs[3];
            g3[0] = idxs[4]; g3[1] = idxs[5]; g3[2] = idxs[6]; g3[3] = idxs[7];

            g0[1] = lds0 + (unsigned)(base * APITCH * 4);   // LDS dest for this chunk
            g1[4] = (unsigned)nvalid;                        // tile_dim1 = #valid indices

            asm volatile("tensor_load_to_lds %0, %1, %2, %3"
                         :: "s"(g0), "s"(g1), "s"(g2), "s"(g3)
                         : "memory");
        }
    }

    // ---- meanwhile: stage W (64x32) and a (64) into LDS ----
    for (int e = tid; e < DIM * HALF_DIM; e += 128)
        sW[(e >> 5) * WPITCH + (e & 31)] = Wg[e];
    if (tid < DIM) sa[tid] = ag[tid];

    // ---- zero pad rows 51..63 of A ----
    {
        const int sub = tid >> 6;
        const int col = tid & 63;
        for (int r = NNB + 1 + sub; r < ROWS; r += 2)
            sA[r * APITCH + col] = 0.0f;
    }

    if (wave == 0) __builtin_amdgcn_s_wait_tensorcnt(0);  // gather complete
    __syncthreads();

    // ---- renorm rows 0..50 (parallel: 102 partial sums -> scales -> apply) ----
    if (tid < 2 * (NNB + 1)) {
        const int r = tid >> 1, hf = tid & 1;
        const int base = r * APITCH + hf * 32;
        float ss = 0.f;
        #pragma unroll
        for (int k = 0; k < 32; ++k) { float v = sA[base + k]; ss += v * v; }
        sPart[tid] = ss;
    }
    __syncthreads();
    if (tid <= NNB) {
        const float n = sqrtf(sPart[2 * tid] + sPart[2 * tid + 1]);
        sScale[tid] = (n > 1.0f) ? (1.0f / (n + 1e-7f)) : 1.0f;
    }
    __syncthreads();
    {
        const int sub = tid >> 6;
        const int col = tid & 63;
        for (int r0 = 0; r0 < 52; r0 += 2) {
            const int r = r0 + sub;
            if (r <= NNB) sA[r * APITCH + col] *= sScale[r];
        }
    }
    __syncthreads();

    // ---- NW(64x32) = A(64x64) @ W(64x32) via v_wmma_f32_16x16x4_f32 ----
    {
        const int half    = lane >> 4;
        const int mn      = lane & 15;
        const int rowbase = (wave * 16 + mn) * APITCH;
        for (int ct = 0; ct < 2; ++ct) {
            v8f acc = {};
            const int cb = ct * 16 + mn;
            for (int kk = 0; kk < 16; ++kk) {
                const int k0 = kk * 4 + 2 * half;
                v2f av, bv;
                av.x = sA[rowbase + k0];
                av.y = sA[rowbase + k0 + 1];
                bv.x = sW[k0 * WPITCH + cb];
                bv.y = sW[(k0 + 1) * WPITCH + cb];
                acc = __builtin_amdgcn_wmma_f32_16x16x4_f32(
                        false, av, false, bv, (short)0, acc, false, false);
            }
            #pragma unroll
            for (int v = 0; v < 8; ++v) {
                const int row = wave * 16 + v + 8 * half;
                sNW[row * NWPITCH + ct * 16 + mn] = acc[v];
            }
        }
    }
    __syncthreads();

    // ---- attention logits: rows 0..49 use a2=a[32:], row 50 (tw) uses a1=a[:32]
    if (tid <= NNB) {
        const float* arow = (tid < NNB) ? (sa + HALF_DIM) : sa;
        float acc = 0.f;
        for (int c = 0; c < HALF_DIM; ++c)
            acc += sNW[tid * NWPITCH + c] * arow[c];
        sE[tid] = acc;
    }
    __syncthreads();

    // ---- leaky-ReLU + softmax over 50 neighbors (wave 0, wave32 shuffles) ----
    if (wave == 0) {
        const float tdot = sE[NNB];
        float v0 = -1e30f, v1 = -1e30f;
        const int l2 = lane + 32;
        if (lane < NNB) { float e = sE[lane] + tdot; v0 = (e >= 0.f) ? e : 0.2f * e; }
        if (l2   < NNB) { float e = sE[l2]   + tdot; v1 = (e >= 0.f) ? e : 0.2f * e; }
        float m = fmaxf(v0, v1);
        for (int off = 16; off > 0; off >>= 1) m = fmaxf(m, __shfl_xor(m, off, 32));
        float e0 = (lane < NNB) ? __expf(v0 - m) : 0.f;
        float e1 = (l2   < NNB) ? __expf(v1 - m) : 0.f;
        float s = e0 + e1;
        for (int off = 16; off > 0; off >>= 1) s += __shfl_xor(s, off, 32);
        const float inv = 1.0f / s;
        if (lane < NNB) sE[lane] = e0 * inv;
        if (l2   < NNB) sE[l2]   = e1 * inv;
    }
    __syncthreads();

    // ---- h[c] = sum_r aij[r]*NW[r][c], split over 4 waves then combined ----
    {
        const int c = tid & 31;
        const int q = tid >> 5;
        int rbeg = q * 13;
        int rend = rbeg + 13; if (rend > NNB) rend = NNB;
        float acc = 0.f;
        for (int r = rbeg; r < rend; ++r)
            acc += sE[r] * sNW[r * NWPITCH + c];
        sHP[q * HALF_DIM + c] = acc;
    }
    __syncthreads();
    if (tid < HALF_DIM)
        sH[tid] = sHP[tid] + sHP[HALF_DIM + tid] + sHP[2 * HALF_DIM + tid]
                + sHP[3 * HALF_DIM + tid];
    __syncthreads();

    // ---- renormed user dot with items=[h,h], sigmoid (wave 0) ----
    if (wave == 0) {
        const long uid = u[b];
        const float u0 = user_emb[uid * DIM + lane];
        const float u1 = user_emb[uid * DIM + 32 + lane];
        float ss = u0 * u0 + u1 * u1;
        for (int off = 16; off > 0; off >>= 1) ss += __shfl_xor(ss, off, 32);
        const float n = sqrtf(ss);
        const float scale = (n > 1.0f) ? (1.0f / (n + 1e-7f)) : 1.0f;
        float p = scale * (u0 + u1) * sH[lane];
        for (int off = 16; off > 0; off >>= 1) p += __shfl_xor(p, off, 32);
        if (lane == 0) out[b] = 1.0f / (1.0f + __expf(-p));
    }
}

extern "C" void kernel_launch(void* const* d_in, const int* in_sizes, int n_in,
                              void* d_out, int out_size, void* d_ws, size_t ws_size,
                              hipStream_t stream) {
    (void)n_in; (void)d_ws; (void)ws_size; (void)out_size;
    const int*   u          = (const int*)d_in[0];
    const int*   itm        = (const int*)d_in[1];
    const int*   neighbors  = (const int*)d_in[2];
    const float* user_emb   = (const float*)d_in[3];
    const float* entity_emb = (const float*)d_in[4];
    const float* W          = (const float*)d_in[5];
    const float* a          = (const float*)d_in[6];
    float*       out        = (float*)d_out;

    const int batch      = in_sizes[0];              // BATCH = 16384
    const int n_entities = in_sizes[4] / DIM;        // 200000
    gat4rec_fused<<<batch, 128, 0, stream>>>(u, itm, neighbors, user_emb,
                                             entity_emb, W, a, out, n_entities);
}